// GATv2ConvLayer_50199577756220
// MI455X (gfx1250) — compile-verified
//
#include <hip/hip_runtime.h>
#include <math.h>

#define NN   50000
#define EE   800000
#define DIN  128
#define HC   128     // H*C
#define NH   4
#define ED   16

typedef float v2f __attribute__((ext_vector_type(2)));
typedef float v8f __attribute__((ext_vector_type(8)));

// ---- order-preserving float<->uint encoding so u32 atomicMax == float max ----
__device__ __forceinline__ unsigned fenc(float f) {
  unsigned u = __float_as_uint(f);
  return (u >> 31) ? ~u : (u | 0x80000000u);
}
__device__ __forceinline__ float fdec(unsigned e) {
  unsigned u = (e >> 31) ? (e & 0x7fffffffu) : ~e;
  return __uint_as_float(u);
}

// ============================ K0: init scratch ===============================
__global__ __launch_bounds__(256) void k_init(float* __restrict__ acc,
                                              float* __restrict__ denom,
                                              unsigned* __restrict__ maxenc) {
  int t = blockIdx.x * 256 + threadIdx.x;
  if (t < NN * HC) acc[t] = 0.0f;
  if (t < NN * NH) { denom[t] = 0.0f; maxenc[t] = 0x007FFFFFu; /* enc(-inf) */ }
}

// ================= K1: xl = x@Wl + bl ; xr = x@Wr + br (WMMA fp32) ==========
__global__ __launch_bounds__(256) void k_gemm(const float* __restrict__ x,
                                              const float* __restrict__ Wl,
                                              const float* __restrict__ bl,
                                              const float* __restrict__ Wr,
                                              const float* __restrict__ br,
                                              float* __restrict__ xl,
                                              float* __restrict__ xr) {
  __shared__ float xs[16 * DIN];
  const int tid = threadIdx.x;
  const float* W    = blockIdx.y ? Wr : Wl;
  const float* bias = blockIdx.y ? br : bl;
  float*       Y    = blockIdx.y ? xr : xl;
  const int rowBase = blockIdx.x * 16;

  // stage 16x128 x-tile to LDS (2048 floats, 8 per thread, coalesced)
#pragma unroll
  for (int i = 0; i < 8; ++i)
    xs[tid + 256 * i] = x[(size_t)rowBase * DIN + tid + 256 * i];
  __syncthreads();

  const int wave = tid >> 5;          // 8 waves -> 8 column tiles of 16
  const int lane = tid & 31;
  const int lr   = lane & 15;
  const bool hi  = lane >= 16;
  const int col  = wave * 16 + lr;

  v8f c = {0.f, 0.f, 0.f, 0.f, 0.f, 0.f, 0.f, 0.f};
#if __has_builtin(__builtin_amdgcn_wmma_f32_16x16x4_f32)
#pragma unroll 8
  for (int t = 0; t < 32; ++t) {
    const int k0 = 4 * t + (hi ? 2 : 0);
    v2f a, b;
    // A 16x4: VGPR0 = K {k0}, VGPR1 = K {k0+1}; lanes 0-15 / 16-31 split K pairs
    a.x = xs[lr * DIN + k0];
    a.y = xs[lr * DIN + k0 + 1];
    // B 4x16: rows of W striped across lanes, same K split as A
    b.x = W[(size_t)k0 * HC + col];
    b.y = W[(size_t)(k0 + 1) * HC + col];
    c = __builtin_amdgcn_wmma_f32_16x16x4_f32(false, a, false, b, (short)0, c,
                                              false, false);
  }
#else
  // scalar fallback producing identical per-lane C/D layout
  for (int r = 0; r < 8; ++r) {
    const int m = r + (hi ? 8 : 0);
    float s = 0.f;
    for (int k = 0; k < DIN; ++k) s += xs[m * DIN + k] * W[(size_t)k * HC + col];
    c[r] = s;
  }
#endif
  const float bb = bias[col];
#pragma unroll
  for (int r = 0; r < 8; ++r) {
    const int m = r + (hi ? 8 : 0);           // C/D layout: VGPR r -> rows r, r+8
    Y[(size_t)(rowBase + m) * HC + col] = c[r] + bb;
  }
}

// =========== K2: per-edge logits + segment-max (1 edge per wave) ============
__global__ __launch_bounds__(256) void k_edge_logits(
    const long long* __restrict__ ei, const float* __restrict__ eattr,
    const float* __restrict__ We, const float* __restrict__ att,
    const float* __restrict__ xl, const float* __restrict__ xr,
    float* __restrict__ logits, unsigned* __restrict__ maxenc) {
  __shared__ float sWe[ED * HC];
  __shared__ float sAtt[HC];
  const int tid = threadIdx.x;
#pragma unroll
  for (int i = 0; i < 8; ++i) sWe[tid + 256 * i] = We[tid + 256 * i];
  if (tid < HC) sAtt[tid] = att[tid];
  __syncthreads();

  const int wave = tid >> 5;
  const int lane = tid & 31;
  const long long e = (long long)blockIdx.x * 8 + wave;
  if (e >= EE) return;

  const int s = (int)ei[e];        // src
  const int d = (int)ei[EE + e];   // dst
  const int h = lane >> 3;         // head (32 ch per head, 4 ch per lane)
  const int ch = lane * 4;

  const float4 xlv = ((const float4*)(xl + (size_t)s * HC))[lane];
  const float4 xrv = ((const float4*)(xr + (size_t)d * HC))[lane];
  const float4 ea0 = ((const float4*)(eattr + (size_t)e * ED))[0];
  const float4 ea1 = ((const float4*)(eattr + (size_t)e * ED))[1];
  const float4 ea2 = ((const float4*)(eattr + (size_t)e * ED))[2];
  const float4 ea3 = ((const float4*)(eattr + (size_t)e * ED))[3];

  const float xin[4] = {xlv.x + xrv.x, xlv.y + xrv.y, xlv.z + xrv.z, xlv.w + xrv.w};
  float p = 0.f;
#pragma unroll
  for (int j = 0; j < 4; ++j) {
    const float* wc = sWe + (ch + j);
    float ef = 0.f;
    ef = fmaf(ea0.x, wc[0 * HC], ef);  ef = fmaf(ea0.y, wc[1 * HC], ef);
    ef = fmaf(ea0.z, wc[2 * HC], ef);  ef = fmaf(ea0.w, wc[3 * HC], ef);
    ef = fmaf(ea1.x, wc[4 * HC], ef);  ef = fmaf(ea1.y, wc[5 * HC], ef);
    ef = fmaf(ea1.z, wc[6 * HC], ef);  ef = fmaf(ea1.w, wc[7 * HC], ef);
    ef = fmaf(ea2.x, wc[8 * HC], ef);  ef = fmaf(ea2.y, wc[9 * HC], ef);
    ef = fmaf(ea2.z, wc[10 * HC], ef); ef = fmaf(ea2.w, wc[11 * HC], ef);
    ef = fmaf(ea3.x, wc[12 * HC], ef); ef = fmaf(ea3.y, wc[13 * HC], ef);
    ef = fmaf(ea3.z, wc[14 * HC], ef); ef = fmaf(ea3.w, wc[15 * HC], ef);
    float v = xin[j] + ef;
    v = v > 0.f ? v : 0.2f * v;          // leaky_relu(0.2)
    p = fmaf(sAtt[ch + j], v, p);
  }
  // reduce partial dot over the 8 lanes of this head
  p += __shfl_xor(p, 1, 8);
  p += __shfl_xor(p, 2, 8);
  p += __shfl_xor(p, 4, 8);
  if ((lane & 7) == 0) {
    logits[e * NH + h] = p;
    atomicMax(maxenc + (size_t)d * NH + h, fenc(p));
  }
}

// ===== K3: ex = exp(logit - m[dst]); scatter denom and ex*xl[src] ===========
__global__ __launch_bounds__(256) void k_edge_scatter(
    const long long* __restrict__ ei, const float* __restrict__ xl,
    const float* __restrict__ logits, const unsigned* __restrict__ maxenc,
    float* __restrict__ denom, float* __restrict__ acc) {
  const int tid = threadIdx.x;
  const int wave = tid >> 5;
  const int lane = tid & 31;
  const long long e = (long long)blockIdx.x * 8 + wave;
  if (e >= EE) return;

  const int s = (int)ei[e];
  const int d = (int)ei[EE + e];
  const int h = lane >> 3;

  float m = fdec(maxenc[(size_t)d * NH + h]);
  if (!(m > -3.0e38f)) m = 0.f;                 // mirror where(isfinite(m), m, 0)
  const float ex = expf(logits[e * NH + h] - m);
  if ((lane & 7) == 0) atomicAdd(denom + (size_t)d * NH + h, ex);

  const float4 xlv = ((const float4*)(xl + (size_t)s * HC))[lane];
  float* ap = acc + (size_t)d * HC + lane * 4;
  atomicAdd(ap + 0, ex * xlv.x);
  atomicAdd(ap + 1, ex * xlv.y);
  atomicAdd(ap + 2, ex * xlv.z);
  atomicAdd(ap + 3, ex * xlv.w);
}

// ====== K4: out = LN(gelu(acc/denom + bias) + x) * gamma + beta =============
__global__ __launch_bounds__(256) void k_node_epilogue(
    const float* __restrict__ x, const float* __restrict__ acc,
    const float* __restrict__ denom, const float* __restrict__ bias,
    const float* __restrict__ gamma, const float* __restrict__ beta,
    float* __restrict__ out) {
  const int tid = threadIdx.x;
  const int wave = tid >> 5;
  const int lane = tid & 31;
  const int n = blockIdx.x * 8 + wave;
  if (n >= NN) return;
  const int h = lane >> 3;

  const float inv = 1.0f / (denom[(size_t)n * NH + h] + 1e-16f);
  const float4 av  = ((const float4*)(acc + (size_t)n * HC))[lane];
  const float4 xv  = ((const float4*)(x + (size_t)n * HC))[lane];
  const float4 bv  = ((const float4*)bias)[lane];
  const float4 gv  = ((const float4*)gamma)[lane];
  const float4 btv = ((const float4*)beta)[lane];

  float o[4] = {av.x * inv + bv.x, av.y * inv + bv.y,
                av.z * inv + bv.z, av.w * inv + bv.w};
  const float xr[4] = {xv.x, xv.y, xv.z, xv.w};
  float s1 = 0.f, s2 = 0.f;
#pragma unroll
  for (int j = 0; j < 4; ++j) {
    float g = o[j];
    g = 0.5f * g * (1.0f + erff(g * 0.70710678118654752440f));  // exact GELU
    g += xr[j];                                                  // residual
    o[j] = g;
    s1 += g;
    s2 += g * g;
  }
#pragma unroll
  for (int off = 1; off < 32; off <<= 1) {
    s1 += __shfl_xor(s1, off);
    s2 += __shfl_xor(s2, off);
  }
  const float mean = s1 * (1.0f / HC);
  float var = s2 * (1.0f / HC) - mean * mean;
  const float rstd = rsqrtf(fmaxf(var, 0.f) + 1e-5f);
  const float g4[4] = {gv.x, gv.y, gv.z, gv.w};
  const float b4[4] = {btv.x, btv.y, btv.z, btv.w};
  float4 r;
  r.x = (o[0] - mean) * rstd * g4[0] + b4[0];
  r.y = (o[1] - mean) * rstd * g4[1] + b4[1];
  r.z = (o[2] - mean) * rstd * g4[2] + b4[2];
  r.w = (o[3] - mean) * rstd * g4[3] + b4[3];
  ((float4*)(out + (size_t)n * HC))[lane] = r;
}

// ============================== launcher ====================================
extern "C" void kernel_launch(void* const* d_in, const int* in_sizes, int n_in,
                              void* d_out, int out_size, void* d_ws, size_t ws_size,
                              hipStream_t stream) {
  const float*     x     = (const float*)d_in[0];
  const long long* ei    = (const long long*)d_in[1];
  const float*     eattr = (const float*)d_in[2];
  const float*     Wl    = (const float*)d_in[3];
  const float*     bl    = (const float*)d_in[4];
  const float*     Wr    = (const float*)d_in[5];
  const float*     br    = (const float*)d_in[6];
  const float*     We    = (const float*)d_in[7];
  const float*     att   = (const float*)d_in[8];
  const float*     bias  = (const float*)d_in[9];
  const float*     gamma = (const float*)d_in[10];
  const float*     beta  = (const float*)d_in[11];
  float*           out   = (float*)d_out;

  float* ws = (float*)d_ws;
  float*    xl     = ws;                                    // N*128
  float*    xr     = xl + (size_t)NN * HC;                  // N*128
  float*    logits = xr + (size_t)NN * HC;                  // E*4
  unsigned* maxenc = (unsigned*)(logits + (size_t)EE * NH); // N*4
  float*    denom  = (float*)(maxenc + (size_t)NN * NH);    // N*4
  float*    acc    = denom + (size_t)NN * NH;               // N*128

  k_init<<<(NN * HC + 255) / 256, 256, 0, stream>>>(acc, denom, maxenc);
  k_gemm<<<dim3(NN / 16, 2), 256, 0, stream>>>(x, Wl, bl, Wr, br, xl, xr);
  k_edge_logits<<<EE / 8, 256, 0, stream>>>(ei, eattr, We, att, xl, xr, logits, maxenc);
  k_edge_scatter<<<EE / 8, 256, 0, stream>>>(ei, xl, logits, maxenc, denom, acc);
  k_node_epilogue<<<(NN + 7) / 8, 256, 0, stream>>>(x, acc, denom, bias, gamma, beta, out);
}